// ELM_AE_SpatialAutomatic_Pool2d_2671469658355
// MI455X (gfx1250) — compile-verified
//
#include <hip/hip_runtime.h>
#include <hip/hip_bf16.h>

// Problem geometry (from reference): x is [32, 512, 56, 56] fp32.
#define B   32
#define Z   512
#define WH  (56 * 56)            // 3136 elements per row
#define ROW_VEC (WH / 4)         // 784 float4 per row
#define BATCH_ELEMS ((size_t)Z * WH)      // 1,605,632
#define BATCH_VEC   (BATCH_ELEMS / 4)     // 401,408 float4
#define ROWS (B * Z)             // 16384 output rows

// ---------------------------------------------------------------------------
// Kernel 0: zero the 32 per-batch max slots in workspace.
// ---------------------------------------------------------------------------
__global__ void k_init(unsigned* __restrict__ bmax) {
    if (threadIdx.x < B) bmax[threadIdx.x] = 0u;
}

// ---------------------------------------------------------------------------
// Kernel 1: per-batch max(|x|).  grid = (blocksPerBatch, B), block = 256.
// Streams the batch with b128 loads, reduces within wave32 via shfl_xor,
// one LDS step across the 8 waves, then a single global_atomic_max_u32
// per block (uint compare == float compare for non-negative floats).
// ---------------------------------------------------------------------------
__global__ void k_batch_max(const float* __restrict__ x,
                            unsigned* __restrict__ bmax) {
    const int b = blockIdx.y;
    const float4* __restrict__ xv =
        (const float4*)(x + (size_t)b * BATCH_ELEMS);

    const int stride = gridDim.x * blockDim.x;
    float m = 0.0f;
    for (int i = blockIdx.x * blockDim.x + threadIdx.x; i < (int)BATCH_VEC; i += stride) {
        // Deepen the pipeline: prefetch next tile (lowers to global_prefetch_b8).
        if (i + stride < (int)BATCH_VEC) __builtin_prefetch(&xv[i + stride], 0, 0);
        float4 v = xv[i];
        float a = fmaxf(fabsf(v.x), fabsf(v.y));
        float c = fmaxf(fabsf(v.z), fabsf(v.w));
        m = fmaxf(m, fmaxf(a, c));
    }

    // wave32 max-reduction (gfx1250 is wave32-only)
    #pragma unroll
    for (int off = 16; off > 0; off >>= 1)
        m = fmaxf(m, __shfl_xor(m, off, 32));

    __shared__ float sm[8];
    const int lane = threadIdx.x & 31;
    const int wid  = threadIdx.x >> 5;
    if (lane == 0) sm[wid] = m;
    __syncthreads();
    if (threadIdx.x == 0) {
        float mm = sm[0];
        #pragma unroll
        for (int i = 1; i < 8; ++i) mm = fmaxf(mm, sm[i]);
        atomicMax(&bmax[b], __float_as_uint(mm));   // |x| >= 0: uint order == float order
    }
}

// ---------------------------------------------------------------------------
// Kernel 2: per-row entropy.  One wave per row, 8 rows per 256-thread block.
// Blocks walk rows in REVERSE order so this pass re-reads the data pass 1
// most recently pushed through the 192 MB L2 (working set is 205 MB), serving
// most of the second stream from L2 instead of HBM.
//
//   p   = fma(x, 1/m, 1)            in [0, 2]
//   ent = log2(s) - sum(p*log2 p)/s,   s = sum(p)
// which equals -sum(pk*log2 pk) with pk = p/s (reference), with 0*log0 := 0.
// ---------------------------------------------------------------------------
__global__ void k_entropy(const float* __restrict__ x,
                          const unsigned* __restrict__ bmax,
                          float* __restrict__ out) {
    const int lane = threadIdx.x & 31;
    const int wid  = threadIdx.x >> 5;

    const int blk = (int)gridDim.x - 1 - (int)blockIdx.x;   // reverse for L2 reuse
    const int row = blk * 8 + wid;                           // [0, ROWS)
    const int b   = row >> 9;                                // row / Z (Z = 512)

    const float rcpm = 1.0f / __uint_as_float(bmax[b]);
    const float4* __restrict__ xv = (const float4*)(x + (size_t)row * WH);

    float s = 0.0f, t = 0.0f;
    #pragma unroll 4
    for (int k = lane; k < ROW_VEC; k += 32) {
        float4 v = xv[k];
        {
            float p = fmaf(v.x, rcpm, 1.0f);
            s += p; t += (p > 0.0f) ? p * __log2f(p) : 0.0f;
        }
        {
            float p = fmaf(v.y, rcpm, 1.0f);
            s += p; t += (p > 0.0f) ? p * __log2f(p) : 0.0f;
        }
        {
            float p = fmaf(v.z, rcpm, 1.0f);
            s += p; t += (p > 0.0f) ? p * __log2f(p) : 0.0f;
        }
        {
            float p = fmaf(v.w, rcpm, 1.0f);
            s += p; t += (p > 0.0f) ? p * __log2f(p) : 0.0f;
        }
    }

    // wave32 sum-reduction; never leaves the wave (no LDS, no barrier).
    #pragma unroll
    for (int off = 16; off > 0; off >>= 1) {
        s += __shfl_xor(s, off, 32);
        t += __shfl_xor(t, off, 32);
    }

    if (lane == 0) out[row] = __log2f(s) - t / s;
}

// ---------------------------------------------------------------------------
extern "C" void kernel_launch(void* const* d_in, const int* in_sizes, int n_in,
                              void* d_out, int out_size, void* d_ws, size_t ws_size,
                              hipStream_t stream) {
    (void)in_sizes; (void)n_in; (void)out_size; (void)ws_size;

    const float* x    = (const float*)d_in[0];
    float*       out  = (float*)d_out;
    unsigned*    bmax = (unsigned*)d_ws;       // 32 uints of scratch

    // Pass 0: clear per-batch max slots.
    k_init<<<1, 64, 0, stream>>>(bmax);

    // Pass 1: per-batch max |x|.  128 blocks x 32 batches, 256 threads.
    dim3 g1(128, B, 1);
    k_batch_max<<<g1, 256, 0, stream>>>(x, bmax);

    // Pass 2: per-row entropy.  8 rows (waves) per block -> 2048 blocks.
    k_entropy<<<ROWS / 8, 256, 0, stream>>>(x, bmax, out);
}